// ConvMatrix2d_74801150427166
// MI455X (gfx1250) — compile-verified
//
#include <hip/hip_runtime.h>

// CDNA5 / gfx1250: wave32, WMMA f32 16x16x4 (K=4 matches the m-contraction).
typedef float v2f __attribute__((ext_vector_type(2)));
typedef float v4f __attribute__((ext_vector_type(4)));
typedef float v8f __attribute__((ext_vector_type(8)));

#define BB    4
#define DD    32
#define WIN   14
#define CH    17          // 16 pose floats + 1 activation
#define OO    32
#define WOUT  12
#define XSLAB (WIN*WIN*CH)   // 3332 floats per (b,d) slab
#define WTILE (OO*16)        // 512 floats per (d,k1,k2) weight tile

__global__ __launch_bounds__(256)
void capsule_votes_wmma(const float* __restrict__ x,
                        const float* __restrict__ wgt,
                        float* __restrict__ out) {
  __shared__ float lds_x[XSLAB];
  __shared__ __align__(16) float lds_w[WTILE];

  const int blk = blockIdx.x;        // 0..1151 = (b, d*9 + k1*3 + k2)
  const int b   = blk / 288;
  const int rem = blk % 288;         // Bkk index = d*9 + k1*3 + k2
  const int d   = rem / 9;
  const int kk  = rem % 9;
  const int kA  = kk / 3;            // k1
  const int kB  = kk % 3;            // k2

  // Stage x[b,d,:,:,:] slab and weight[d,kA,kB,:,:,:] tile into LDS.
  const float* gx = x + (size_t)(b * DD + d) * XSLAB;
  for (int i = threadIdx.x; i < XSLAB; i += 256) lds_x[i] = gx[i];
  const float* gw = wgt + (size_t)rem * WTILE;
  for (int i = threadIdx.x; i < WTILE; i += 256) lds_w[i] = gw[i];
  __syncthreads();

  const int wave = threadIdx.x >> 5;
  const int lane = threadIdx.x & 31;
  const int half = lane >> 4;        // K-half for A/B operands
  const int ln   = lane & 15;        // M (pose) for A, N (t) for B/D
  const int p    = ln >> 2;          // N = p*4 + q  (output vote indices)
  const int q    = ln & 3;
  const bool mx  = (q == half * 2);      // B vgpr0 row = K = half*2
  const bool my  = (q == half * 2 + 1);  // B vgpr1 row = K = half*2+1

  // out[b, rem, Cidx, t], Cidx = o*144 + u*12 + v, flat stride 17.
  float* obase = out + (size_t)(b * 288 + rem) * 4608u * 17u;

  // Each wave owns 4 consecutive o's; loop pose-groups outer so the A
  // operand (pose matrices) and act are loaded from LDS once per group
  // and reused across the 4 WMMA chains.
  const int o0 = wave * 4;

  for (int g = 0; g < 9; ++g) {      // 9 pose groups of 16 poses
    const int P    = g * 16 + ln;    // linear pose = u*12 + v
    const int u    = P / WOUT;
    const int v    = P % WOUT;
    const int cell = ((v + kA) * WIN + (u + kB)) * CH;
    const int co   = cell + half * 2;

    v2f a0, a1, a2, a3;
    a0.x = lds_x[co + 0];  a0.y = lds_x[co + 1];
    a1.x = lds_x[co + 4];  a1.y = lds_x[co + 5];
    a2.x = lds_x[co + 8];  a2.y = lds_x[co + 9];
    a3.x = lds_x[co + 12]; a3.y = lds_x[co + 13];
    const float act = lds_x[cell + 16];

    float* gp = obase + (size_t)(g * 16) * 17u;

#pragma unroll
    for (int oi = 0; oi < 4; ++oi) {
      const int o = o0 + oi;

      // W_o[p, m=0..3]: one aligned float4 per lane.
      const v4f wv = *(const v4f*)&lds_w[o * 16 + p * 4];

      // B chunk kc is block-diagonal: B[(m=kc,q'),(p,q)] = W[p,kc]*(q'==q).
      v2f b0, b1, b2, b3;
      b0.x = mx ? wv.x : 0.f;  b0.y = my ? wv.x : 0.f;
      b1.x = mx ? wv.y : 0.f;  b1.y = my ? wv.y : 0.f;
      b2.x = mx ? wv.z : 0.f;  b2.y = my ? wv.z : 0.f;
      b3.x = mx ? wv.w : 0.f;  b3.y = my ? wv.w : 0.f;

      v8f acc = {};
      acc = __builtin_amdgcn_wmma_f32_16x16x4_f32(false, a0, false, b0, (short)0, acc, false, false);
      acc = __builtin_amdgcn_wmma_f32_16x16x4_f32(false, a1, false, b1, (short)0, acc, false, false);
      acc = __builtin_amdgcn_wmma_f32_16x16x4_f32(false, a2, false, b2, (short)0, acc, false, false);
      acc = __builtin_amdgcn_wmma_f32_16x16x4_f32(false, a3, false, b3, (short)0, acc, false, false);

      // D[M=pose, N=t]: vgpr j -> pose j (+8 for upper half), lane ln -> t.
      // Each store = two contiguous 64B runs of the final [pose,t] layout.
      float* op = gp + (size_t)(o * 144 + half * 8) * 17u + ln;
#pragma unroll
      for (int j = 0; j < 8; ++j) op[(size_t)j * 17u] = acc[j];

      // Activation at t=16, one dword per (o, pose); lower half covers
      // the group's 16 poses.
      if (half == 0) gp[(size_t)(o * 144 + ln) * 17u + 16u] = act;
    }
  }
}

extern "C" void kernel_launch(void* const* d_in, const int* in_sizes, int n_in,
                              void* d_out, int out_size, void* d_ws, size_t ws_size,
                              hipStream_t stream) {
  const float* x = (const float*)d_in[0];
  const float* w = (const float*)d_in[1];
  float* out    = (float*)d_out;
  dim3 grid(BB * 288);   // one workgroup per (b, d, k1, k2)
  dim3 block(256);       // 8 wave32
  hipLaunchKernelGGL(capsule_votes_wmma, grid, block, 0, stream, x, w, out);
}